// RNN_20332375179300
// MI455X (gfx1250) — compile-verified
//
#include <hip/hip_runtime.h>
#include <math.h>

// Problem constants (match reference)
#define B_TOTAL 4096
#define T_LEN   512
#define INP     6
#define HID     8

typedef float v2f __attribute__((ext_vector_type(2)));
typedef float v8f __attribute__((ext_vector_type(8)));

// D = A(16x4) * B(4x16) + C(16x16), all f32.
static __device__ __forceinline__ v8f wmma4(v2f a, v2f b, v8f c) {
  return __builtin_amdgcn_wmma_f32_16x16x4_f32(false, a, false, b, (short)0, c,
                                               false, false);
}

// Branch-free tanh: prefer the gfx1250 hardware v_tanh_f32.
static __device__ __forceinline__ float htanh(float x) {
#if __has_builtin(__builtin_amdgcn_tanhf)
  return __builtin_amdgcn_tanhf(x);
#else
  // tanh(x) = 1 - 2/(e^{2x}+1); exp2(2*log2(e)*x) via HW v_exp_f32.
  float t = __builtin_amdgcn_exp2f(x * 2.8853900817779268f);
  return 1.0f - 2.0f * __builtin_amdgcn_rcpf(t + 1.0f);
#endif
}

static __device__ __forceinline__ v8f tanh8(v8f d) {
  v8f h;
#pragma unroll
  for (int v = 0; v < 8; ++v) h[v] = htanh(d[v]);
  return h;
}

// D-layout -> B-layout: upper half-wave of D holds rows 8..15 which are
// exactly zero, so a half-wave swap + add yields K=2hi+{0,1} (b0) and
// K=4+2hi+{0,1} (b1) in each lane.
static __device__ __forceinline__ void packB(const v8f& h, v2f& b0, v2f& b1) {
  b0.x = h[0] + __shfl_xor(h[2], 16, 32);
  b0.y = h[1] + __shfl_xor(h[3], 16, 32);
  b1.x = h[4] + __shfl_xor(h[6], 16, 32);
  b1.y = h[5] + __shfl_xor(h[7], 16, 32);
}

// A operand (16x4 tile of W, K = kbase..kbase+3): lane m (=lane&15) holds row
// m; VGPR0 = K(kbase+2*hi), VGPR1 = K(kbase+2*hi+1). Zero-padded.
static __device__ __forceinline__ v2f load_A(const float* __restrict__ W,
                                             int ncols, int kbase, int m,
                                             int hi) {
  int k0 = kbase + 2 * hi;
  v2f a;
  a.x = (m < HID && k0 < ncols) ? W[m * ncols + k0] : 0.0f;
  a.y = (m < HID && (k0 + 1) < ncols) ? W[m * ncols + k0 + 1] : 0.0f;
  return a;
}

__global__ __launch_bounds__(128) void rnn2_tanh_wmma(
    const float* __restrict__ x,
    const float* __restrict__ Wih0, const float* __restrict__ Whh0,
    const float* __restrict__ bih0, const float* __restrict__ bhh0,
    const float* __restrict__ Wih1, const float* __restrict__ Whh1,
    const float* __restrict__ bih1, const float* __restrict__ bhh1,
    const float* __restrict__ Wout, const float* __restrict__ bout_p,
    float* __restrict__ out) {
  const int lane = threadIdx.x & 31;
  const int wid = (blockIdx.x * blockDim.x + threadIdx.x) >> 5;
  const int n = lane & 15;   // batch column in tile (also row m for A)
  const int hi = lane >> 4;  // half-wave selector for K pairs
  const int bb = wid * 16;   // base batch of this wave's tile

  // ---- constant A operands (weights) ----
  const v2f a_ih0_0 = load_A(Wih0, INP, 0, n, hi);
  const v2f a_ih0_1 = load_A(Wih0, INP, 4, n, hi);
  const v2f a_hh0_0 = load_A(Whh0, HID, 0, n, hi);
  const v2f a_hh0_1 = load_A(Whh0, HID, 4, n, hi);
  const v2f a_ih1_0 = load_A(Wih1, HID, 0, n, hi);
  const v2f a_ih1_1 = load_A(Wih1, HID, 4, n, hi);
  const v2f a_hh1_0 = load_A(Whh1, HID, 0, n, hi);
  const v2f a_hh1_1 = load_A(Whh1, HID, 4, n, hi);

  // ---- bias C matrices (rows 0-7 on lanes 0-15; zero elsewhere) ----
  v8f c0b, c1b;
#pragma unroll
  for (int v = 0; v < 8; ++v) {
    c0b[v] = (hi == 0) ? (bih0[v] + bhh0[v]) : 0.0f;
    c1b[v] = (hi == 0) ? (bih1[v] + bhh1[v]) : 0.0f;
  }

  float wo[8];
#pragma unroll
  for (int v = 0; v < 8; ++v) wo[v] = Wout[v];
  const float bo = bout_p[0];
  const float ximask = (hi == 0) ? 1.0f : 0.0f;  // zeroes K=6,7 pad lanes

  // recurrent-state B operands (h0 = 0)
  v2f h1b0 = (v2f){0.f, 0.f}, h1b1 = (v2f){0.f, 0.f};
  v2f h2b0 = (v2f){0.f, 0.f}, h2b1 = (v2f){0.f, 0.f};

  const float* xp = x + (size_t)(bb + n) * T_LEN * INP;
  float* op = out + (size_t)(bb + n) * T_LEN;
  const int off0 = 2 * hi;            // K = 2hi, 2hi+1 of block 0
  const int off1 = hi ? 0 : 4;        // hi lanes: harmless duplicate load

  // x B-operand loader (branch-free; hi half zeroed for K block 1 pad)
  auto loadX = [&](int t, v2f& b0, v2f& b1) {
    const float* xq = xp + (size_t)t * INP;
    b0 = *(const v2f*)(xq + off0);
    v2f r = *(const v2f*)(xq + off1);
    b1.x = r.x * ximask;
    b1.y = r.y * ximask;
  };

  // Prologue: dx = bias + Wih0*x(0)^T ; xb holds x(1).
  v2f xb0, xb1;
  loadX(0, xb0, xb1);
  v8f dx = wmma4(a_ih0_0, xb0, c0b);
  dx = wmma4(a_ih0_1, xb1, dx);
  loadX(1, xb0, xb1);

  float4 ob;

#pragma unroll 4
  for (int t = 0; t < T_LEN; ++t) {
    // Off-critical-path: x-projection for step t+1 (uses preloaded xb).
    v8f dxn = wmma4(a_ih0_0, xb0, c0b);
    dxn = wmma4(a_ih0_1, xb1, dxn);

    // Off-critical-path: layer-1 recurrent part (depends only on h2[t-1]).
    v8f erec = wmma4(a_hh1_0, h2b0, c1b);
    erec = wmma4(a_hh1_1, h2b1, erec);

    // Critical path: layer-0 recurrence -> tanh -> pack.
    v8f d = wmma4(a_hh0_0, h1b0, dx);
    d = wmma4(a_hh0_1, h1b1, d);
    v8f h1 = tanh8(d);
    packB(h1, h1b0, h1b1);

    // Layer 1 input part -> tanh -> pack.
    v8f e = wmma4(a_ih1_0, h1b0, erec);
    e = wmma4(a_ih1_1, h1b1, e);
    v8f h2 = tanh8(e);
    packB(h2, h2b0, h2b1);

    // Output projection; buffer 4 t's, one 16B store per batch row.
    float o = bo;
#pragma unroll
    for (int v = 0; v < 8; ++v) o = fmaf(wo[v], h2[v], o);

    const int ph = t & 3;
    if (ph == 0) {
      ob.x = o;
      __builtin_prefetch(xp + (size_t)(t + 16) * INP, 0, 1);  // ~384B ahead
    } else if (ph == 1) {
      ob.y = o;
    } else if (ph == 2) {
      ob.z = o;
    } else {
      ob.w = o;
      if (lane < 16) *(float4*)(op + t - 3) = ob;
    }

    // Advance pipeline: dx for t+1, fetch x for t+2 (uniform clamp, no branch).
    dx = dxn;
    int t2 = t + 2;
    t2 = t2 < T_LEN ? t2 : T_LEN - 1;
    loadX(t2, xb0, xb1);
  }
}

extern "C" void kernel_launch(void* const* d_in, const int* in_sizes, int n_in,
                              void* d_out, int out_size, void* d_ws,
                              size_t ws_size, hipStream_t stream) {
  (void)in_sizes; (void)n_in; (void)out_size; (void)d_ws; (void)ws_size;
  const float* x    = (const float*)d_in[0];
  const float* Wih0 = (const float*)d_in[1];
  const float* Whh0 = (const float*)d_in[2];
  const float* bih0 = (const float*)d_in[3];
  const float* bhh0 = (const float*)d_in[4];
  const float* Wih1 = (const float*)d_in[5];
  const float* Whh1 = (const float*)d_in[6];
  const float* bih1 = (const float*)d_in[7];
  const float* bhh1 = (const float*)d_in[8];
  const float* Wout = (const float*)d_in[9];
  const float* bout = (const float*)d_in[10];
  float* out = (float*)d_out;

  // 4096 batches / 16 per wave = 256 waves; 4 waves (128 threads) per block.
  dim3 grid(B_TOTAL / 16 / 4);
  dim3 block(128);
  rnn2_tanh_wmma<<<grid, block, 0, stream>>>(x, Wih0, Whh0, bih0, bhh0, Wih1,
                                             Whh1, bih1, bhh1, Wout, bout, out);
}